// MO_RIS_90391881712210
// MI455X (gfx1250) — compile-verified
//
#include <hip/hip_runtime.h>
#include <math.h>

typedef float v2f __attribute__((ext_vector_type(2)));
typedef float v8f __attribute__((ext_vector_type(8)));

// ---------------- complex helpers (float2, interleaved complex64) ----------
__device__ __forceinline__ float2 cmk(float a, float b){ float2 r; r.x=a; r.y=b; return r; }
__device__ __forceinline__ float2 cadd(float2 a, float2 b){ return cmk(a.x+b.x, a.y+b.y); }
__device__ __forceinline__ float2 csub(float2 a, float2 b){ return cmk(a.x-b.x, a.y-b.y); }
__device__ __forceinline__ float2 cmul(float2 a, float2 b){ return cmk(a.x*b.x - a.y*b.y, a.x*b.y + a.y*b.x); }
__device__ __forceinline__ float2 cconj(float2 a){ return cmk(a.x, -a.y); }
__device__ __forceinline__ float2 cscale(float2 a, float s){ return cmk(a.x*s, a.y*s); }
__device__ __forceinline__ float2 cinv(float2 a){ float d = a.x*a.x + a.y*a.y; return cmk(a.x/d, -a.y/d); }

// 4x4 complex Gauss-Jordan inverse (inputs are HPD -> no pivoting needed)
__device__ void cinv4(float2 a[4][4], float2 v[4][4]){
  for (int i=0;i<4;i++) for (int j=0;j<4;j++) v[i][j] = cmk(i==j?1.f:0.f, 0.f);
  for (int c=0;c<4;c++){
    float2 ip = cinv(a[c][c]);
    for (int j=0;j<4;j++){ a[c][j] = cmul(a[c][j], ip); v[c][j] = cmul(v[c][j], ip); }
    for (int r=0;r<4;r++){
      if (r==c) continue;
      float2 f = a[r][c];
      for (int j=0;j<4;j++){ a[r][j] = csub(a[r][j], cmul(f, a[c][j])); v[r][j] = csub(v[r][j], cmul(f, v[c][j])); }
    }
  }
}

// ---------------- kernel 0: phi[b,p] = diag(Phi_in) ------------------------
__global__ void k_init_phi(const float2* __restrict__ Phic, float2* __restrict__ phic){
  int id = blockIdx.x*256 + threadIdx.x;          // 16384 = 256*64
  int b = id >> 6, p = id & 63;
  phic[id] = Phic[((size_t)(b*64 + p))*64 + p];
}

// ---------------- kernel 1: Gh[k,b,p,s] = sum_{r,f} conj(H2[b,r,p,k]) Wrf[b,r,f] Wbb[k,b,f,s]
__global__ __launch_bounds__(1024) void k_gh(const float2* __restrict__ H2c,
                                             const float2* __restrict__ Wrfc,
                                             const float2* __restrict__ Wbbc,
                                             float2* __restrict__ Ghc){
  __shared__ float2 sWrf[64];    // [r*4+f]
  __shared__ float2 sWbb[256];   // [k*16 + f*4+s]
  const int b = blockIdx.x;
  const int tid = threadIdx.x;   // tid = p*16 + k  -> coalesced H2 reads
  if (tid < 64)  sWrf[tid] = Wrfc[(size_t)b*64 + tid];
  if (tid < 256) { int k = tid >> 4; sWbb[tid] = Wbbc[((size_t)k*256 + b)*16 + (tid & 15)]; }
  __syncthreads();
  const int p = tid >> 4;
  const int k = tid & 15;
  float2 W2[4] = {cmk(0,0),cmk(0,0),cmk(0,0),cmk(0,0)};
  for (int r=0;r<16;++r){
    float2 h2c = cconj(H2c[(((size_t)b*16 + r)*64 + p)*16 + k]);
    for (int f=0;f<4;++f) W2[f] = cadd(W2[f], cmul(h2c, sWrf[r*4+f]));
  }
  for (int s=0;s<4;++s){
    float2 g = cmk(0,0);
    for (int f=0;f<4;++f) g = cadd(g, cmul(W2[f], sWbb[k*16 + f*4 + s]));
    Ghc[(((size_t)k*256 + b)*64 + p)*4 + s] = g;
  }
}

// ---------------- kernel 2 (WMMA): D[k,b,p,f] = sum_t H1[b,p,t,k] * Frf[b,t,f]
// block = (b, p-tile of 16), 128 threads (4 waves, 4 k each).
// LDS stage of H1[b,p0:p0+16,:,:] is one contiguous 128KB global read.
#define ROWSTRIDE 2052   // 2048 floats per p-row + 4 pad (bank spread, float4 aligned)
__global__ __launch_bounds__(128) void k_dmat(const float* __restrict__ H1f,
                                              const float2* __restrict__ Frfc,
                                              float2* __restrict__ Dc){
  __shared__ __align__(16) float sH1[16*ROWSTRIDE]; // ~128KB (of 320KB LDS)
  __shared__ float sB[64*16];                       // B = [t, f_r(4)|f_i(4)|pad(8)]
  __shared__ float sPQ[4*2*256];                    // per-wave P/Q 16x16 scratch
  const int tid = threadIdx.x;
  const int bid = blockIdx.x;
  const int b  = bid >> 2;
  const int p0 = (bid & 3) << 4;
  // ---- stage H1 tile (fully coalesced float4 stream) ----
  const float4* src4 = (const float4*)H1f + (size_t)(b*64 + p0) * 512;
  for (int i = tid; i < 8192; i += 128){
    int row = i >> 9, col = i & 511;
    ((float4*)(sH1 + row*ROWSTRIDE))[col] = src4[i];
  }
  // ---- build B matrix from Frf[b] ----
  if (tid < 64){
    int t = tid;
    for (int f=0; f<4; ++f){
      float2 v = Frfc[((size_t)b*64 + t)*4 + f];
      sB[t*16 + f]     = v.x;
      sB[t*16 + 4 + f] = v.y;
    }
    for (int j=8;j<16;++j) sB[t*16+j] = 0.f;
  }
  __syncthreads();
  const int w    = tid >> 5;
  const int lane = tid & 31;
  const int half = lane >> 4;
  const int pl   = lane & 15;
  float* sP = sPQ + w*512;
  float* sQ = sP + 256;
  for (int kk=0; kk<4; ++kk){
    int k = w*4 + kk;
    v8f accP = {0,0,0,0,0,0,0,0};
    v8f accQ = {0,0,0,0,0,0,0,0};
    for (int t0=0; t0<64; t0+=4){
      // A 16x4 f32 fragment: vgpr v holds K = t0 + v + 2*half (ISA 7.12.2)
      int tb = t0 + 2*half;
      int f0 = pl*ROWSTRIDE + (tb*16 + k)*2;
      v2f ar, ai, bv;
      ar.x = sH1[f0];      ai.x = sH1[f0+1];
      ar.y = sH1[f0+32];   ai.y = sH1[f0+33];       // t+1 -> +16 complex
      // B 4x16 fragment: vgpr v holds row K = t0 + v + 2*half, col N = pl
      bv.x = sB[tb*16 + pl];
      bv.y = sB[(tb+1)*16 + pl];
      accP = __builtin_amdgcn_wmma_f32_16x16x4_f32(false, ar, false, bv, (short)0, accP, false, false);
      accQ = __builtin_amdgcn_wmma_f32_16x16x4_f32(false, ai, false, bv, (short)0, accQ, false, false);
    }
    // C frag: vgpr j holds M = j + 8*half, N = pl (ISA 7.12.2) -> spill to LDS
    for (int j=0;j<8;++j){
      int m = j + 8*half;
      sP[m*16 + pl] = accP[j];
      sQ[m*16 + pl] = accQ[j];
    }
    // recombine complex: D_r = P[:,f] - Q[:,f+4]; D_i = P[:,f+4] + Q[:,f]
    for (int q = lane*2; q < lane*2+2; ++q){
      int p = q >> 2, f = q & 3;
      float Dr = sP[p*16+f]   - sQ[p*16+f+4];
      float Di = sP[p*16+f+4] + sQ[p*16+f];
      Dc[(((size_t)k*256 + b)*64 + (p0+p))*4 + f] = cmk(Dr, Di);
    }
  }
}

// ---------------- kernel 3: Lam_inv (4096 independent 4x4 complex inverses)
__global__ void k_laminv(const float2* __restrict__ Lamc, float2* __restrict__ Lic){
  int id = blockIdx.x*256 + threadIdx.x;
  if (id >= 4096) return;
  float2 a[4][4], v[4][4];
  for (int i=0;i<4;i++) for (int j=0;j<4;j++) a[i][j] = Lamc[(size_t)id*16 + i*4 + j];
  cinv4(a, v);
  for (int i=0;i<4;i++) for (int j=0;j<4;j++) Lic[(size_t)id*16 + i*4 + j] = v[i][j];
}

// ---------------- kernel 4: per-(k,b) iteration partials -> egk diag -------
__global__ __launch_bounds__(64) void k_partials(const float2* __restrict__ phic,
                                                 const float2* __restrict__ Ghc,
                                                 const float2* __restrict__ Dc,
                                                 const float2* __restrict__ Lic,
                                                 const float2* __restrict__ Betac,
                                                 float2* __restrict__ egkc){
  __shared__ float red[32*64];
  __shared__ float2 sT[17];
  const int bid = blockIdx.x;
  const int k = bid >> 8, b = bid & 255;
  const int p = threadIdx.x;
  const size_t base = (((size_t)k*256 + b)*64 + p)*4;
  float2 phi_p = phic[b*64 + p];
  float2 Ghp[4], Dp[4], Ms[4];
  for (int s=0;s<4;++s) Ghp[s] = Ghc[base + s];
  for (int f=0;f<4;++f) Dp[f]  = Dc[base + f];
  for (int s=0;s<4;++s) Ms[s] = cmul(cconj(Ghp[s]), phi_p);     // M[s,p]
  // E[s][f] = sum_p M[s,p] * D[p,f]  (deterministic tree reduction)
  for (int s=0;s<4;++s) for (int f=0;f<4;++f){
    float2 c = cmul(Ms[s], Dp[f]);
    red[((s*4+f)*2  )*64 + p] = c.x;
    red[((s*4+f)*2+1)*64 + p] = c.y;
  }
  __syncthreads();
  for (int off=32; off>0; off>>=1){
    if (p < off) for (int j=0;j<32;++j) red[j*64+p] += red[j*64+p+off];
    __syncthreads();
  }
  if (p == 0){
    float2 E[4][4];
    for (int s=0;s<4;++s) for (int f=0;f<4;++f)
      E[s][f] = cmk(red[((s*4+f)*2)*64], red[((s*4+f)*2+1)*64]);
    float2 ib = cinv(Betac[k*256 + b]);
    float2 Qm[4][4];
    for (int s=0;s<4;++s) for (int t=0;t<4;++t){       // Q = Lam_inv + E E^H / Beta
      float2 acc = cmk(0,0);
      for (int f=0;f<4;++f) acc = cadd(acc, cmul(E[s][f], cconj(E[t][f])));
      Qm[s][t] = cadd(Lic[((size_t)k*256 + b)*16 + s*4 + t], cmul(ib, acc));
    }
    float2 Qi[4][4]; cinv4(Qm, Qi);
    float2 Qi2[4][4];
    for (int s=0;s<4;++s) for (int t=0;t<4;++t){
      float2 acc = cmk(0,0);
      for (int u=0;u<4;++u) acc = cadd(acc, cmul(Qi[s][u], Qi[u][t]));
      Qi2[s][t] = acc;
    }
    for (int s=0;s<4;++s) for (int f=0;f<4;++f){       // T = Qi^2 * E
      float2 acc = cmk(0,0);
      for (int u=0;u<4;++u) acc = cadd(acc, cmul(Qi2[s][u], E[u][f]));
      sT[s*4+f] = acc;
    }
    sT[16] = ib;
  }
  __syncthreads();
  // diag(eg_k)[p] = sum_{s,f} Gh[p,s] T[s,f] conj(D[p,f]) / Beta
  float2 acc = cmk(0,0);
  for (int s=0;s<4;++s){
    float2 inner = cmk(0,0);
    for (int f=0;f<4;++f) inner = cadd(inner, cmul(sT[s*4+f], cconj(Dp[f])));
    acc = cadd(acc, cmul(Ghp[s], inner));
  }
  egkc[((size_t)k*256 + b)*64 + p] = cmul(acc, sT[16]);
}

// ---------------- kernel 5: fused step-net + manifold update ---------------
__global__ __launch_bounds__(64) void k_update(float2* __restrict__ phic,
                                               const float2* __restrict__ egkc,
                                               const float* __restrict__ bng, const float* __restrict__ bnb,
                                               const float* __restrict__ bnm, const float* __restrict__ bnv,
                                               const float* __restrict__ dw,  const float* __restrict__ db,
                                               int iter){
  __shared__ float red[64];
  __shared__ float sval[2];
  const int b = blockIdx.x;
  const int p = threadIdx.x;
  const float* g  = bng + iter*128;
  const float* be = bnb + iter*128;
  const float* mu = bnm + iter*128;
  const float* va = bnv + iter*128;
  const float* wv = dw  + iter*128;
  float2 phi_p = phic[b*64 + p];
  // step net: BN(inference, eps=1e-3) + Dense(1) + leaky_relu(0.1)
  float hr = (phi_p.x - mu[p])    * rsqrtf(va[p]    + 1e-3f) * g[p]    + be[p];
  float hi = (phi_p.y - mu[64+p]) * rsqrtf(va[64+p] + 1e-3f) * g[64+p] + be[64+p];
  red[p] = hr * wv[p] + hi * wv[64+p];
  __syncthreads();
  for (int off=32; off>0; off>>=1){ if (p<off) red[p] += red[p+off]; __syncthreads(); }
  if (p==0){ float s = red[0] + db[iter]; sval[0] = (s < 0.f) ? 0.1f*s : s; }
  __syncthreads();
  // eg = -mean_k(eg_k); Riemannian projection on unit-modulus manifold
  float2 eg = cmk(0,0);
  for (int k=0;k<16;++k) eg = cadd(eg, egkc[((size_t)k*256 + b)*64 + p]);
  eg = cscale(eg, -1.f/16.f);
  float proj = eg.x*phi_p.x + eg.y*phi_p.y;       // real(eg * conj(phi))
  float2 rg = csub(eg, cscale(phi_p, proj));
  red[p] = rg.x*rg.x + rg.y*rg.y;
  __syncthreads();
  for (int off=32; off>0; off>>=1){ if (p<off) red[p] += red[p+off]; __syncthreads(); }
  if (p==0) sval[1] = sqrtf(red[0]);
  __syncthreads();
  float coef = sval[0] / sval[1];
  float2 pn = csub(phi_p, cscale(rg, coef));
  float mag = sqrtf(pn.x*pn.x + pn.y*pn.y);
  float den = fmaxf(mag - 1.f, 0.f) + 1.f;
  phic[b*64 + p] = cscale(pn, 1.f/den);
}

// ---------------- kernel 6: materialize full diagonal Phi ------------------
__global__ void k_out(const float2* __restrict__ phic, float2* __restrict__ out){
  const int b = blockIdx.x;
  for (int idx = threadIdx.x; idx < 4096; idx += 256){
    int p = idx >> 6, q = idx & 63;
    float2 v = (p == q) ? phic[b*64 + p] : cmk(0.f, 0.f);
    out[(size_t)b*4096 + idx] = v;
  }
}

extern "C" void kernel_launch(void* const* d_in, const int* in_sizes, int n_in,
                              void* d_out, int out_size, void* d_ws, size_t ws_size,
                              hipStream_t stream) {
  const float2* Frfc = (const float2*)d_in[0];
  const float2* Wrfc = (const float2*)d_in[1];
  const float2* Wbbc = (const float2*)d_in[2];
  const float2* Lamc = (const float2*)d_in[3];
  const float*  H1f  = (const float*)d_in[4];
  const float2* H2c  = (const float2*)d_in[5];
  const float2* Phic = (const float2*)d_in[6];
  const float2* Betac= (const float2*)d_in[7];
  const float* bng = (const float*)d_in[8];
  const float* bnb = (const float*)d_in[9];
  const float* bnm = (const float*)d_in[10];
  const float* bnv = (const float*)d_in[11];
  const float* dw  = (const float*)d_in[12];
  const float* db  = (const float*)d_in[13];

  float2* Ghc  = (float2*)d_ws;            // 16*256*64*4 = 1,048,576 cplx (8MB)
  float2* Dc   = Ghc  + 1048576;           // 1,048,576 cplx (8MB)
  float2* Lic  = Dc   + 1048576;           // 16*256*16  =    65,536 cplx
  float2* phic = Lic  + 65536;             // 256*64     =    16,384 cplx
  float2* egkc = phic + 16384;             // 16*256*64  =   262,144 cplx

  k_init_phi<<<64, 256, 0, stream>>>(Phic, phic);
  k_gh      <<<256, 1024, 0, stream>>>(H2c, Wrfc, Wbbc, Ghc);
  k_dmat    <<<1024, 128, 0, stream>>>(H1f, Frfc, Dc);
  k_laminv  <<<16, 256, 0, stream>>>(Lamc, Lic);
  for (int i=0;i<4;++i){
    k_partials<<<4096, 64, 0, stream>>>(phic, Ghc, Dc, Lic, Betac, egkc);
    k_update  <<<256, 64, 0, stream>>>(phic, egkc, bng, bnb, bnm, bnv, dw, db, i);
  }
  k_out<<<256, 256, 0, stream>>>(phic, (float2*)d_out);
  (void)in_sizes; (void)n_in; (void)out_size; (void)ws_size;
}